// RelationalGraphConv_21766894256811
// MI455X (gfx1250) — compile-verified
//
#include <hip/hip_runtime.h>
#include <hip/hip_bf16.h>

// ---- problem constants (match reference) ----
#define N_NODES 100000
#define N_EDGES 1600000
#define IN_DIM  32
#define OUT_DIM 64
#define N_REL   8
#define EPS     1e-10f

#define K_TOT     (N_REL * IN_DIM + IN_DIM) // 288 = 256 (update) + 32 (x)
#define AS_STRIDE 292                       // 288 + 4 pad -> conflict-free LDS banks
#define M_TILE    32                        // nodes per block (2 WMMA row tiles)

typedef float v2f __attribute__((ext_vector_type(2)));
typedef float v8f __attribute__((ext_vector_type(8)));

// -----------------------------------------------------------------------------
// Phase 1: edge scatter. One wave32 per edge; lane == feature (IN_DIM == 32).
//   num[(dst*8+rel)*32 + lane] += w * x[src*32 + lane]
//   den[dst*8+rel]             += w        (lane 0)
// Coalesced 128B row reads; atomics resolve in L2 (num = 102MB < 192MB L2).
// -----------------------------------------------------------------------------
__global__ void rgcn_scatter(const float* __restrict__ x,
                             const int*   __restrict__ edge_list,
                             const float* __restrict__ edge_weight,
                             float* __restrict__ num,
                             float* __restrict__ den)
{
    const int lane   = threadIdx.x & 31;
    const int wave   = (blockIdx.x * blockDim.x + threadIdx.x) >> 5;
    const int nwaves = (gridDim.x * blockDim.x) >> 5;

    for (int e = wave; e < N_EDGES; e += nwaves) {
        const int src = edge_list[e * 3 + 0];
        const int dst = edge_list[e * 3 + 1];
        const int rel = edge_list[e * 3 + 2];
        const float w = edge_weight[e];
        const int seg = dst * N_REL + rel;

        const float v = x[(size_t)src * IN_DIM + lane];
        atomicAdd(&num[(size_t)seg * IN_DIM + lane], v * w);
        if (lane == 0) atomicAdd(&den[seg], w);
    }
}

// -----------------------------------------------------------------------------
// Phase 2: out = relu( [update | x] @ [W_linear ; W_self] + (b_linear+b_self) )
// Block = 128 threads (4 waves), M-tile = 32 nodes. Wave w owns output columns
// [16w, 16w+16) and TWO row tiles (M 0..15 and 16..31): each B fragment is
// loaded once and feeds two V_WMMA_F32_16X16X4_F32 ops.
// -----------------------------------------------------------------------------
__global__ void rgcn_gemm(const float* __restrict__ x,
                          const float* __restrict__ num,
                          const float* __restrict__ den,
                          const float* __restrict__ Wl,   // 256 x 64
                          const float* __restrict__ bl,   // 64
                          const float* __restrict__ Ws,   // 32 x 64
                          const float* __restrict__ bs,   // 64
                          float* __restrict__ out)        // N x 64
{
    __shared__ float invden[M_TILE * N_REL];      // 1/(den+eps) per (node, rel)
    __shared__ float As[M_TILE * AS_STRIDE];      // 32 x 288 A-tile

    const int tid   = threadIdx.x;
    const int node0 = blockIdx.x * M_TILE;        // 100000 / 32 = 3125 exact

    // ---- stage 0: reciprocal denominators (256 entries, 128 threads) ----
    for (int i = tid; i < M_TILE * N_REL; i += 128) {
        const int m = i >> 3;
        const int r = i & 7;
        invden[i] = 1.0f / (den[(size_t)(node0 + m) * N_REL + r] + EPS);
    }
    __syncthreads();

    // ---- stage 1: build A-tile in LDS: update (k<256) then x (k>=256) ----
    for (int idx = tid; idx < M_TILE * K_TOT; idx += 128) {
        const int m = idx / K_TOT;
        const int k = idx - m * K_TOT;
        const int node = node0 + m;
        float v;
        if (k < 256) {
            v = num[(size_t)node * 256 + k] * invden[m * N_REL + (k >> 5)];
        } else {
            v = x[(size_t)node * IN_DIM + (k - 256)];
        }
        As[m * AS_STRIDE + k] = v;
    }
    __syncthreads();

    // ---- stage 2: WMMA K-loop, two row tiles share each B fragment ----
    const int lane  = tid & 31;
    const int wavew = tid >> 5;                   // 0..3 -> N tile
    const int half  = lane >> 4;                  // 0 or 1 (lane group)
    const int l15   = lane & 15;
    const int col   = wavew * 16 + l15;           // global output column

    const float biasv = bl[col] + bs[col];
    v8f c0, c1;
    #pragma unroll
    for (int r = 0; r < 8; ++r) { c0[r] = biasv; c1[r] = biasv; }

    const float* As0 = &As[l15 * AS_STRIDE];             // rows 0..15
    const float* As1 = &As[(l15 + 16) * AS_STRIDE];      // rows 16..31

    // K = 0..255 : update part against W_linear
    #pragma unroll 8
    for (int k0 = 0; k0 < 256; k0 += 4) {
        const int kr = k0 + half * 2;
        v2f a0, a1, b;
        a0.x = As0[kr];  a0.y = As0[kr + 1];
        a1.x = As1[kr];  a1.y = As1[kr + 1];
        b.x  = Wl[(size_t)kr * OUT_DIM + col];
        b.y  = Wl[(size_t)(kr + 1) * OUT_DIM + col];
        c0 = __builtin_amdgcn_wmma_f32_16x16x4_f32(false, a0, false, b,
                                                   (short)0, c0, false, false);
        c1 = __builtin_amdgcn_wmma_f32_16x16x4_f32(false, a1, false, b,
                                                   (short)0, c1, false, false);
    }
    // K = 256..287 : self part against W_self
    #pragma unroll
    for (int k0 = 256; k0 < K_TOT; k0 += 4) {
        const int kr = k0 + half * 2;
        const int ks = kr - 256;
        v2f a0, a1, b;
        a0.x = As0[kr];  a0.y = As0[kr + 1];
        a1.x = As1[kr];  a1.y = As1[kr + 1];
        b.x  = Ws[(size_t)ks * OUT_DIM + col];
        b.y  = Ws[(size_t)(ks + 1) * OUT_DIM + col];
        c0 = __builtin_amdgcn_wmma_f32_16x16x4_f32(false, a0, false, b,
                                                   (short)0, c0, false, false);
        c1 = __builtin_amdgcn_wmma_f32_16x16x4_f32(false, a1, false, b,
                                                   (short)0, c1, false, false);
    }

    // ---- store with ReLU. C layout: vgpr r -> row (r + 8*half), col l15 ----
    #pragma unroll
    for (int r = 0; r < 8; ++r) {
        const int row0 = node0 + r + half * 8;          // tile 0: rows 0..15
        const int row1 = row0 + 16;                     // tile 1: rows 16..31
        out[(size_t)row0 * OUT_DIM + col] = fmaxf(c0[r], 0.0f);
        out[(size_t)row1 * OUT_DIM + col] = fmaxf(c1[r], 0.0f);
    }
}

// -----------------------------------------------------------------------------
extern "C" void kernel_launch(void* const* d_in, const int* in_sizes, int n_in,
                              void* d_out, int out_size, void* d_ws, size_t ws_size,
                              hipStream_t stream)
{
    const float* x  = (const float*)d_in[0];
    const int*   el = (const int*)  d_in[1];
    const float* ew = (const float*)d_in[2];
    const float* Wl = (const float*)d_in[3];
    const float* bl = (const float*)d_in[4];
    const float* Ws = (const float*)d_in[5];
    const float* bs = (const float*)d_in[6];
    float* out = (float*)d_out;

    // workspace: num [N*256] then den [N*8]
    float* num = (float*)d_ws;
    float* den = num + (size_t)N_NODES * (N_REL * IN_DIM);
    const size_t accum_bytes =
        ((size_t)N_NODES * (N_REL * IN_DIM) + (size_t)N_NODES * N_REL) * sizeof(float);

    hipMemsetAsync(d_ws, 0, accum_bytes, stream);

    rgcn_scatter<<<2048, 256, 0, stream>>>(x, el, ew, num, den);

    rgcn_gemm<<<N_NODES / M_TILE, 128, 0, stream>>>(x, num, den, Wl, bl, Ws, bs, out);
}